// EMARU_3375844294868
// MI455X (gfx1250) — compile-verified
//
#include <hip/hip_runtime.h>
#include <math.h>

// MI455X / gfx1250 fused kernel.
// One workgroup (256 threads = 8 wave32) per bg sample (512 blocks).
// gx (32x1024 fp32) and gated live in LDS (row stride 1036 floats, ==12 mod 64,
// so WMMA B-loads (rows +2) and C/D stores (rows +8) avoid bank conflicts).
// GEMM xh = w1 * gx done with V_WMMA_F32_16X16X4_F32 (fp32 precision; the problem
// is HBM-bound at ~128MB traffic -> ~5.5us @ 23.3TB/s, so fp32 WMMA is ample).

#define CG 32
#define H  1024
#define BG 512
#define STRIDE 1036
#define EPSV 1e-5f
#define THR 0.5f

typedef __attribute__((ext_vector_type(2))) float v2f;
typedef __attribute__((ext_vector_type(8))) float v8f;

// Fast sigmoid: v_exp_f32 + v_rcp_f32 (avoids the IEEE div_scale/div_fixup chain).
__device__ __forceinline__ float sigm(float x) {
  return __builtin_amdgcn_rcpf(1.0f + __expf(-x));
}

// LDS layout (float offsets)
#define OFF_GX    0
#define OFF_GATED (CG*STRIDE)
#define OFF_W1    (2*CG*STRIDE)          // 1024: w1[o][i]
#define OFF_WM    (OFF_W1 + 1024)        // 3072: wm[c][i][ky] = w3[c,i,ky,1]
#define OFF_SIGW  (OFF_WM + 3072)        // 1024: sigmoid(weights[y])
#define OFF_B1    (OFF_SIGW + 1024)
#define OFF_B3    (OFF_B1 + 32)
#define OFF_GAM   (OFF_B3 + 32)
#define OFF_BET   (OFF_GAM + 32)
#define OFF_CS    (OFF_BET + 32)         // channel sums of gx
#define OFF_SW    (OFF_CS + 32)          // sigmoid(xw) per channel
#define OFF_MX2   (OFF_SW + 32)          // mean over h of conv output x2
#define OFF_X11   (OFF_MX2 + 32)
#define OFF_X21   (OFF_X11 + 32)
#define OFF_SG    (OFF_X21 + 32)         // sum(gated)
#define OFF_SG2   (OFF_SG + 32)          // sum(gated^2)
#define OFF_COEF  (OFF_SG2 + 32)         // x21[c]*a_c
#define OFF_TRM   (OFF_COEF + 32)        // x21[c]*(beta_c - a_c*mu_c)
#define OFF_V     (OFF_TRM + 32)         // 96: v[i][ky] = sum_c x11[c]*wm[c][i][ky]
#define OFF_SC    (OFF_V + 96)           // [0]=WC  [1]=cb
#define SMEM_FLOATS (OFF_SC + 8)

__global__ void __launch_bounds__(256) fused_emaru_kernel(
    const float* __restrict__ x,  const float* __restrict__ w1,
    const float* __restrict__ b1, const float* __restrict__ w3,
    const float* __restrict__ b3, const float* __restrict__ gamma,
    const float* __restrict__ beta, float* __restrict__ out)
{
  extern __shared__ float sm[];
  const int tid = threadIdx.x;
  const int blk = blockIdx.x;
  const float* gx = x + (size_t)blk * (CG * H);

  // ---------------- P0: stage gx + params into LDS ----------------
  #pragma unroll
  for (int k = 0; k < 32; ++k) {            // 8192 float4 total
    int id = tid + 256 * k;
    int c  = id >> 8;                        // 256 float4 per row
    int y4 = (id & 255) << 2;
    float4 v = ((const float4*)gx)[id];
    *((float4*)&sm[OFF_GX + c * STRIDE + y4]) = v;
  }
  {
    float4 v = ((const float4*)w1)[tid];     // 1024 floats
    *((float4*)&sm[OFF_W1 + tid * 4]) = v;
  }
  #pragma unroll
  for (int k = 0; k < 12; ++k) {             // middle column of 3x3 filters
    int e = tid + 256 * k;                   // 3072 elements
    int o = e / 96; int rem = e - o * 96; int i = rem / 3; int ky = rem - i * 3;
    sm[OFF_WM + e] = w3[o * 288 + i * 9 + ky * 3 + 1];
  }
  if (tid < 32) {
    sm[OFF_B1 + tid]  = b1[tid];
    sm[OFF_B3 + tid]  = b3[tid];
    sm[OFF_GAM + tid] = gamma[tid];
    sm[OFF_BET + tid] = beta[tid];
    sm[OFF_CS + tid]  = 0.0f;
    sm[OFF_SG + tid]  = 0.0f;
    sm[OFF_SG2 + tid] = 0.0f;
  }
  __syncthreads();

  // ---------------- P1: channel sums of gx ----------------
  {
    int c = tid >> 3, part = tid & 7;
    const float* row = &sm[OFF_GX + c * STRIDE + part * 128];
    float s = 0.f;
    #pragma unroll 8
    for (int j = 0; j < 128; j += 4) {
      float4 v = *((const float4*)(row + j));
      s += v.x + v.y + v.z + v.w;
    }
    atomicAdd(&sm[OFF_CS + c], s);
  }
  __syncthreads();

  // ---------------- P2a: per-channel gate sigma(xw), mean(x2) ----------------
  if (tid < 32) {
    const int o = tid;
    const float invH = 1.0f / (float)H;
    float acc = sm[OFF_B1 + o];
    for (int i = 0; i < 32; ++i)
      acc += sm[OFF_W1 + o * 32 + i] * (sm[OFF_CS + i] * invH);
    sm[OFF_SW + o] = sigm(acc);
    // mean over h of conv(gx): needs only channel sums + edge samples
    float m = sm[OFF_B3 + o];
    for (int i = 0; i < 32; ++i) {
      float S  = sm[OFF_CS + i];
      float gf = sm[OFF_GX + i * STRIDE + 0];
      float gl = sm[OFF_GX + i * STRIDE + (H - 1)];
      m += invH * ( sm[OFF_WM + o * 96 + i * 3 + 0] * (S - gl)
                  + sm[OFF_WM + o * 96 + i * 3 + 1] *  S
                  + sm[OFF_WM + o * 96 + i * 3 + 2] * (S - gf) );
    }
    sm[OFF_MX2 + o] = m;
  }
  __syncthreads();

  // ---------------- P2b: softmaxes (tiny, serial) ----------------
  if (tid == 0) {                            // x21 = softmax(mean(x2))
    float mx = -1e30f;
    for (int c = 0; c < 32; ++c) mx = fmaxf(mx, sm[OFF_MX2 + c]);
    float s = 0.f;
    for (int c = 0; c < 32; ++c) { float e = __expf(sm[OFF_MX2 + c] - mx); sm[OFF_X21 + c] = e; s += e; }
    float inv = __builtin_amdgcn_rcpf(s);
    for (int c = 0; c < 32; ++c) sm[OFF_X21 + c] *= inv;
  } else if (tid == 32) {                    // mean(x1) == beta exactly -> x11 = softmax(beta)
    float mx = -1e30f;
    for (int c = 0; c < 32; ++c) mx = fmaxf(mx, sm[OFF_BET + c]);
    float s = 0.f;
    for (int c = 0; c < 32; ++c) { float e = __expf(sm[OFF_BET + c] - mx); sm[OFF_X11 + c] = e; s += e; }
    float inv = __builtin_amdgcn_rcpf(s);
    for (int c = 0; c < 32; ++c) sm[OFF_X11 + c] *= inv;
  }
  __syncthreads();

  // ---------------- P2c: collapsed conv filter v[i][ky], cb ----------------
  if (tid < 96) {
    int i = tid / 3, ky = tid - i * 3;
    float a = 0.f;
    for (int c = 0; c < 32; ++c) a += sm[OFF_X11 + c] * sm[OFF_WM + c * 96 + i * 3 + ky];
    sm[OFF_V + i * 3 + ky] = a;
  }
  if (tid == 128) {
    float cb = 0.f;
    for (int c = 0; c < 32; ++c) cb += sm[OFF_X11 + c] * sm[OFF_B3 + c];
    sm[OFF_SC + 1] = cb;
  }
  __syncthreads();

  // ---------------- P3: WMMA GEMM xh = w1*gx, fused sigmoid gate ----------------
  {
    const int wave = tid >> 5, lane = tid & 31;
    const int lhi = lane >> 4;               // half-wave select
    const int lm  = lane & 15;
    // A fragments (w1, 16x4 fp32 layout): lanes 0-15 K={0,1}, lanes 16-31 K={2,3}
    v2f afrag[2][8];
    #pragma unroll
    for (int mt = 0; mt < 2; ++mt)
      #pragma unroll
      for (int kk = 0; kk < 8; ++kk) {
        int row = mt * 16 + lm;
        int kb  = kk * 4 + (lhi << 1);
        afrag[mt][kk].x = sm[OFF_W1 + row * 32 + kb];
        afrag[mt][kk].y = sm[OFF_W1 + row * 32 + kb + 1];
      }
    // Per-lane epilogue constants: each lane's 16 channels are fixed across N-tiles.
    float b1r[2][8], swr[2][8];
    #pragma unroll
    for (int mt = 0; mt < 2; ++mt)
      #pragma unroll
      for (int vg = 0; vg < 8; ++vg) {
        int o = mt * 16 + vg + (lhi << 3);
        b1r[mt][vg] = sm[OFF_B1 + o];
        swr[mt][vg] = sm[OFF_SW + o];
      }
    for (int nt = 0; nt < 8; ++nt) {
      int n0 = (wave * 8 + nt) * 16;
      v8f acc0 = {0,0,0,0,0,0,0,0};
      v8f acc1 = {0,0,0,0,0,0,0,0};
      #pragma unroll
      for (int kk = 0; kk < 8; ++kk) {
        int kb = kk * 4 + (lhi << 1);
        v2f b;
        b.x = sm[OFF_GX + kb * STRIDE + n0 + lm];
        b.y = sm[OFF_GX + (kb + 1) * STRIDE + n0 + lm];
        acc0 = __builtin_amdgcn_wmma_f32_16x16x4_f32(false, afrag[0][kk], false, b,
                                                     (short)0, acc0, false, false);
        acc1 = __builtin_amdgcn_wmma_f32_16x16x4_f32(false, afrag[1][kk], false, b,
                                                     (short)0, acc1, false, false);
      }
      // epilogue: gated = gx * sigmoid(xh) * sigmoid(xw)
      int y = n0 + lm;
      #pragma unroll
      for (int vg = 0; vg < 8; ++vg) {
        int o0 = vg + (lhi << 3);
        {
          int o = o0;                        // m-tile 0
          float xh = acc0[vg] + b1r[0][vg];
          float g  = sm[OFF_GX + o * STRIDE + y];
          sm[OFF_GATED + o * STRIDE + y] = g * sigm(xh) * swr[0][vg];
        }
        {
          int o = 16 + o0;                   // m-tile 1
          float xh = acc1[vg] + b1r[1][vg];
          float g  = sm[OFF_GX + o * STRIDE + y];
          sm[OFF_GATED + o * STRIDE + y] = g * sigm(xh) * swr[1][vg];
        }
      }
    }
  }
  __syncthreads();

  // ---------------- P4: group-norm stats of gated ----------------
  {
    int c = tid >> 3, part = tid & 7;
    const float* row = &sm[OFF_GATED + c * STRIDE + part * 128];
    float s = 0.f, s2 = 0.f;
    #pragma unroll 8
    for (int j = 0; j < 128; j += 4) {
      float4 v = *((const float4*)(row + j));
      s  += v.x + v.y + v.z + v.w;
      s2 += v.x * v.x + v.y * v.y + v.z * v.z + v.w * v.w;
    }
    atomicAdd(&sm[OFF_SG + c], s);
    atomicAdd(&sm[OFF_SG2 + c], s2);
  }
  __syncthreads();
  if (tid < 32) {
    const int c = tid;
    float mu  = sm[OFF_SG + c] * (1.0f / (float)H);
    float var = sm[OFF_SG2 + c] * (1.0f / (float)H) - mu * mu;
    float a   = sm[OFF_GAM + c] * rsqrtf(var + EPSV);
    sm[OFF_COEF + c] = sm[OFF_X21 + c] * a;
    sm[OFF_TRM + c]  = sm[OFF_X21 + c] * (sm[OFF_BET + c] - a * mu);
  }
  __syncthreads();
  if (tid == 0) {
    float k = sm[OFF_SC + 1];
    for (int c = 0; c < 32; ++c) k += sm[OFF_TRM + c];
    sm[OFF_SC + 0] = k;                      // WC: all constant terms of weights[y]
  }
  __syncthreads();

  // ---------------- P5a: weights[y] -> sigmoid, into LDS ----------------
  {
    const float WC = sm[OFF_SC + 0];
    #pragma unroll
    for (int rep = 0; rep < 4; ++rep) {
      int y = tid + rep * 256;
      float wsum = WC;
      for (int i = 0; i < 32; ++i) {
        float g0 = sm[OFF_GX + i * STRIDE + y];
        float gm = (y > 0)     ? sm[OFF_GX + i * STRIDE + y - 1] : 0.f;
        float gp = (y < H - 1) ? sm[OFF_GX + i * STRIDE + y + 1] : 0.f;
        wsum += sm[OFF_V + i * 3 + 0] * gm + sm[OFF_V + i * 3 + 1] * g0
              + sm[OFF_V + i * 3 + 2] * gp;
        wsum += sm[OFF_COEF + i] * sm[OFF_GATED + i * STRIDE + y];
      }
      sm[OFF_SIGW + y] = sigm(wsum);
    }
  }
  __syncthreads();

  // ---------------- P5b: threshold gate + half-channel swap, coalesced store ----
  {
    float* o = out + (size_t)blk * (CG * H);
    #pragma unroll
    for (int k = 0; k < 32; ++k) {
      int id = tid + 256 * k;
      int c  = id >> 8;
      int y4 = (id & 255) << 2;
      int p  = c ^ 16;                        // partner half-channel
      float4 g  = *((const float4*)&sm[OFF_GX + c * STRIDE + y4]);
      float4 gp = *((const float4*)&sm[OFF_GX + p * STRIDE + y4]);
      float4 s  = *((const float4*)&sm[OFF_SIGW + y4]);
      float4 r;
      r.x = ((g.x * s.x >= THR) ? g.x : 0.f) + ((gp.x * s.x >= THR) ? 0.f : gp.x);
      r.y = ((g.y * s.y >= THR) ? g.y : 0.f) + ((gp.y * s.y >= THR) ? 0.f : gp.y);
      r.z = ((g.z * s.z >= THR) ? g.z : 0.f) + ((gp.z * s.z >= THR) ? 0.f : gp.z);
      r.w = ((g.w * s.w >= THR) ? g.w : 0.f) + ((gp.w * s.w >= THR) ? 0.f : gp.w);
      ((float4*)o)[id] = r;
    }
  }
}

extern "C" void kernel_launch(void* const* d_in, const int* in_sizes, int n_in,
                              void* d_out, int out_size, void* d_ws, size_t ws_size,
                              hipStream_t stream) {
  const float* x     = (const float*)d_in[0];
  const float* w1    = (const float*)d_in[1];
  const float* b1    = (const float*)d_in[2];
  const float* w3    = (const float*)d_in[3];
  const float* b3    = (const float*)d_in[4];
  const float* gamma = (const float*)d_in[5];
  const float* beta  = (const float*)d_in[6];
  float* out = (float*)d_out;

  const size_t smem = (size_t)SMEM_FLOATS * sizeof(float);  // ~281 KB <= 320 KB/WGP
  (void)hipFuncSetAttribute((const void*)fused_emaru_kernel,
                            hipFuncAttributeMaxDynamicSharedMemorySize, (int)smem);
  fused_emaru_kernel<<<BG, 256, smem, stream>>>(x, w1, b1, w3, b3, gamma, beta, out);
}